// RelationalGraphConv_75350906241166
// MI455X (gfx1250) — compile-verified
//
#include <hip/hip_runtime.h>
#include <hip/hip_bf16.h>

// ---------------------------------------------------------------------------
// R-GCN layer, algebraically refactored:
//   s[v]      = sum_{e: out=v} w_e * x[in_e]              (edge scatter, no FLOPs)
//   relw[v,r] = sum_{e: out=v, rel=r} w_e                 (8 scalars per node)
//   out[v]    = relu( (s[v] @ W + wsum[v]*b + relw[v,:] @ rel_emb) / (wsum[v]+eps) )
// GEMM uses exact-f32 V_WMMA_F32_16X16X4_F32 (16x16 tiles, K=4 per issue).
// ---------------------------------------------------------------------------

typedef __attribute__((ext_vector_type(2))) float v2f;
typedef __attribute__((ext_vector_type(8))) float v8f;

#define N_NODE  50000
#define N_EDGE  625000
#define DIM     128
#define NUM_REL 8
#define EPS_F   1e-6f

// ---------------------------------------------------------------------------
// Phase 1: edge scatter.  One 128-thread block per edge so the per-edge
// scalars (node_in/node_out/relation/edge_weight) are block-uniform -> scalar
// (s_load) fetches.  Each thread handles one channel: one global load of x +
// one global_atomic_add_f32 into the accumulator.  Both x (25.6 MB) and the
// accumulator (25.6 MB) are resident in the 192 MB L2.
// ---------------------------------------------------------------------------
__global__ __launch_bounds__(DIM)
void edge_scatter_kernel(const float* __restrict__ x,
                         const int*   __restrict__ node_in,
                         const int*   __restrict__ node_out,
                         const int*   __restrict__ relation,
                         const float* __restrict__ ew,
                         float*       __restrict__ acc,    // [N_NODE, DIM], zeroed
                         float*       __restrict__ relw)   // [N_NODE, NUM_REL], zeroed
{
    const int e = blockIdx.x;          // one block per edge
    const int c = threadIdx.x;         // channel 0..127

    const int   vin  = node_in[e];     // uniform -> scalar load
    const int   vout = node_out[e];
    const float w    = ew[e];

    const float v = x[(size_t)vin * DIM + c];
    atomicAdd(acc + (size_t)vout * DIM + c, v * w);

    if (c == 0) {
        const int r = relation[e];
        atomicAdd(relw + vout * NUM_REL + r, w);
    }
}

// ---------------------------------------------------------------------------
// Phase 2: in-place GEMM  s -> s @ W  using V_WMMA_F32_16X16X4_F32.
// Block = 256 threads = 8 wave32s; block covers 128 rows, each wave a 16-row
// strip producing 8 C-tiles of 16x16.
//
// W is staged in LDS *K-pair interleaved*: element (k, n) lives at
//   Wlds[((k >> 1) * DIM + n) * 2 + (k & 1)]
// so each lane's B fragment (two consecutive K values, one N column) is a
// single 8-byte-aligned ds_load_b64 landing in an even VGPR pair — directly
// consumable by the WMMA without register shuffles.
//
// A-row guard: out-of-range rows are *clamped* (not zeroed).  Garbage A rows
// only corrupt C rows with the same M index, which are out-of-range too and
// never stored — so the A load is unconditional (no per-iteration exec mask).
//
// In-place is safe: a wave reads only its own 16 rows (entire K loop) before
// writing those same 16 rows back.
//
// f32 WMMA fragment layouts (ISA 7.12.2):
//   A 16x4 : lanes 0-15 hold M=lane, VGPR{0,1}=K{0,1}; lanes 16-31: K{2,3}
//   B 4x16 : lanes 0-15 hold N=lane, VGPR{0,1}=K{0,1}; lanes 16-31: K{2,3}
//   C 16x16: lanes 0-15: VGPR r -> (M=r,   N=lane)
//            lanes16-31: VGPR r -> (M=8+r, N=lane-16)
// ---------------------------------------------------------------------------
__global__ __launch_bounds__(256)
void gemm_inplace_kernel(float* __restrict__ sbuf,          // d_out, in/out
                         const float* __restrict__ Wmat)    // [DIM, DIM]
{
    __shared__ float Wlds[DIM * DIM];                       // 64 KB, K-pair packed
    for (int i = threadIdx.x; i < DIM * DIM; i += blockDim.x) {
        const int k = i >> 7;          // source row    (K index)
        const int n = i & (DIM - 1);   // source column (N index)
        Wlds[(((k >> 1) * DIM) + n) * 2 + (k & 1)] = Wmat[i];
    }
    __syncthreads();

    const int waveId = threadIdx.x >> 5;
    const int lane   = threadIdx.x & 31;
    const int half   = lane >> 4;        // 0: K pair {0,1}, 1: K pair {2,3}
    const int l16    = lane & 15;
    const int r0     = blockIdx.x * 128 + waveId * 16;      // strip base row

    v8f c[8] = {};                                          // 8 N-tiles of C

    // Clamped A row: OOB lanes read valid memory; their C rows are never stored.
    const int arow  = r0 + l16;
    const int crow  = (arow < N_NODE) ? arow : (N_NODE - 1);
    const float* __restrict__ aptr = sbuf + (size_t)crow * DIM + 2 * half;

    // B fragment base for this lane: K-pair (2k + half), column l16.
    const v2f* __restrict__ bbase = (const v2f*)Wlds + half * DIM + l16;

#pragma unroll 4
    for (int k = 0; k < DIM / 4; ++k) {
        const v2f a = *(const v2f*)(aptr + k * 4);          // K = 4k+2h .. +1
        const v2f* __restrict__ brow = bbase + (2 * k) * DIM;
#pragma unroll
        for (int t = 0; t < 8; ++t) {
            const v2f b = brow[t * 16];                     // single ds_load_b64
            c[t] = __builtin_amdgcn_wmma_f32_16x16x4_f32(
                       /*neg_a=*/false, a,
                       /*neg_b=*/false, b,
                       /*c_mod=*/(short)0, c[t],
                       /*reuse_a=*/false, /*reuse_b=*/false);
        }
    }

    // Store C tiles back in place.
#pragma unroll
    for (int t = 0; t < 8; ++t) {
#pragma unroll
        for (int r = 0; r < 8; ++r) {
            const int row = r0 + half * 8 + r;
            if (row < N_NODE)
                sbuf[(size_t)row * DIM + t * 16 + l16] = c[t][r];
        }
    }
}

// ---------------------------------------------------------------------------
// Phase 3: elementwise finalize.
//   out[v][c] = relu( (g[v][c] + wsum*b[c] + sum_r relw[v][r]*rel_emb[r][c])
//                     / (wsum + eps) )
// ---------------------------------------------------------------------------
__global__ __launch_bounds__(256)
void finalize_kernel(float* __restrict__ out,               // d_out, in/out
                     const float* __restrict__ relw,        // [N_NODE, NUM_REL]
                     const float* __restrict__ bias,        // [DIM]
                     const float* __restrict__ rel_emb)     // [NUM_REL, DIM]
{
    const int idx = blockIdx.x * blockDim.x + threadIdx.x;
    if (idx >= N_NODE * DIM) return;
    const int v = idx >> 7;        // / DIM
    const int c = idx & (DIM - 1);

    float wsum = 0.f, rc = 0.f;
#pragma unroll
    for (int j = 0; j < NUM_REL; ++j) {
        const float rw = relw[v * NUM_REL + j];
        wsum += rw;
        rc   += rw * rel_emb[j * DIM + c];
    }
    const float val = out[idx];
    const float res = (val + wsum * bias[c] + rc) / (wsum + EPS_F);
    out[idx] = res > 0.f ? res : 0.f;
}

// ---------------------------------------------------------------------------
// Launcher.  d_in order: x, node_in, node_out, relation, edge_weight,
//                        W, b, rel_emb.
// d_out doubles as the scatter accumulator (saves 25.6 MB of workspace);
// d_ws holds relw (50000*8 floats = 1.6 MB).
// ---------------------------------------------------------------------------
extern "C" void kernel_launch(void* const* d_in, const int* in_sizes, int n_in,
                              void* d_out, int out_size, void* d_ws, size_t ws_size,
                              hipStream_t stream) {
    const float* x        = (const float*)d_in[0];
    const int*   node_in  = (const int*)  d_in[1];
    const int*   node_out = (const int*)  d_in[2];
    const int*   relation = (const int*)  d_in[3];
    const float* ew       = (const float*)d_in[4];
    const float* Wmat     = (const float*)d_in[5];
    const float* bias     = (const float*)d_in[6];
    const float* rel_emb  = (const float*)d_in[7];

    float* out  = (float*)d_out;
    float* relw = (float*)d_ws;

    hipMemsetAsync(out,  0, sizeof(float) * (size_t)N_NODE * DIM, stream);
    hipMemsetAsync(relw, 0, sizeof(float) * (size_t)N_NODE * NUM_REL, stream);

    edge_scatter_kernel<<<N_EDGE, DIM, 0, stream>>>(
        x, node_in, node_out, relation, ew, out, relw);

    gemm_inplace_kernel<<<(N_NODE + 127) / 128, 256, 0, stream>>>(out, Wmat);

    finalize_kernel<<<(N_NODE * DIM + 255) / 256, 256, 0, stream>>>(
        out, relw, bias, rel_emb);
}